// GammaCapsGraph_3521873183198
// MI455X (gfx1250) — compile-verified
//
#include <hip/hip_runtime.h>
#include <cmath>

typedef __attribute__((ext_vector_type(2))) float v2f;
typedef __attribute__((ext_vector_type(4))) float v4f;
typedef __attribute__((ext_vector_type(8))) float v8f;

#define D_DIM 128
#define N_DIM 1024
#define B_DIM 32

// ---------------------------------------------------------------------------
// Kernel 1: u_hat[b] = x[b]^T x[b]  (1024x128 * 128x1024 per batch) via
// V_WMMA_F32_16X16X4_F32, fused with the row-norm rescale
//   scale_row = min(1, ||x[:,n]|| / ||u_hat_row||)
// Each block: one batch b, one 16-row M-tile; 8 waves x 8 N-tiles each cover
// the full 16x1024 slab so row norms reduce inside the block.
// ---------------------------------------------------------------------------
__global__ __launch_bounds__(256) void caps_gemm_scale(
    const float* __restrict__ x, float* __restrict__ uhat)
{
    __shared__ float s_un[16];   // ||x[:,n]||^2  (norm over feature dim)
    __shared__ float s_rq[16];   // ||u_hat_row||^2

    const int tid = threadIdx.x;
    const int b   = blockIdx.y;
    const int m0  = blockIdx.x * 16;
    const float* xb = x + (size_t)b * D_DIM * N_DIM;

    if (tid < 16) { s_un[tid] = 0.f; s_rq[tid] = 0.f; }
    __syncthreads();

    // u_norm^2 for the 16 rows of this tile (reduce over d = 0..127)
    {
        const int row = tid & 15, chunk = tid >> 4;
        const int col = m0 + row;
        float p = 0.f;
#pragma unroll
        for (int j = 0; j < 8; ++j) {
            float v = xb[(size_t)(chunk * 8 + j) * N_DIM + col];
            p += v * v;
        }
        atomicAdd(&s_un[row], p);
    }

    const int lane = tid & 31;
    const int wave = tid >> 5;
    const int lm   = lane & 15;      // M / N index inside tile
    const int lh   = lane >> 4;      // half-wave selects K pair (f32 16x16x4)
    const int mcol = m0 + lm;
    const int nbase = wave * 128;    // 8 N-tiles per wave

    v8f acc[8] = {};

    for (int k0 = 0; k0 < D_DIM; k0 += 4) {
        const float* xr0 = xb + (size_t)(k0 + lh * 2) * N_DIM;
        const float* xr1 = xr0 + N_DIM;
        v2f a; a.x = xr0[mcol]; a.y = xr1[mcol];       // A: 16x4 fragment
#pragma unroll
        for (int t = 0; t < 8; ++t) {
            const int nc = nbase + t * 16 + lm;
            v2f bb; bb.x = xr0[nc]; bb.y = xr1[nc];    // B: 4x16 fragment
            acc[t] = __builtin_amdgcn_wmma_f32_16x16x4_f32(
                false, a, false, bb, (short)0, acc[t], false, false);
        }
    }

    // ||u_hat_row||^2 : per-lane partial over this wave's 128 columns,
    // xor-reduce across the 16 lanes of each half-wave, accumulate in LDS.
#pragma unroll
    for (int r = 0; r < 8; ++r) {
        float ss = 0.f;
#pragma unroll
        for (int t = 0; t < 8; ++t) { float v = acc[t][r]; ss += v * v; }
        ss += __shfl_xor(ss, 1, 16);
        ss += __shfl_xor(ss, 2, 16);
        ss += __shfl_xor(ss, 4, 16);
        ss += __shfl_xor(ss, 8, 16);
        if (lm == 0) atomicAdd(&s_rq[r + lh * 8], ss);
    }
    __syncthreads();

    const size_t outbase = ((size_t)b * N_DIM + m0) * N_DIM;
#pragma unroll
    for (int r = 0; r < 8; ++r) {
        const int row = r + lh * 8;      // C/D layout: lanes16-31 hold M=r+8
        const float sc = fminf(1.0f, sqrtf(s_un[row] / s_rq[row]));
#pragma unroll
        for (int t = 0; t < 8; ++t)
            uhat[outbase + (size_t)row * N_DIM + nbase + t * 16 + lm] =
                acc[t][r] * sc;
    }
}

// ---------------------------------------------------------------------------
// Routing pass: one wave per (b,n) row of u_hat (1024 f32, register resident).
//   s = c[b,n]*u + bias[n,:] ;  f = |s|^2 / ((1+|s|^2) sqrt(|s|^2))
// mode 0: c = 1/N, write d[b,n] = ||f*s - u||
// mode 1: c from softmax stats,  write d
// mode 2: c from softmax stats,  overwrite row in place with v = f*s
// ---------------------------------------------------------------------------
__global__ __launch_bounds__(256) void caps_route_pass(
    float* uhat, const float* __restrict__ bias,
    const float* __restrict__ d_in, const float* __restrict__ stats,
    float* __restrict__ d_new, int mode)
{
    const int tid  = threadIdx.x;
    const int lane = tid & 31;
    const int gid  = blockIdx.x * 8 + (tid >> 5);   // global row id
    const int b    = gid >> 10;
    const int n    = gid & (N_DIM - 1);

    float c;
    if (mode == 0) {
        c = 1.0f / (float)N_DIM;
    } else {
        const float t   = stats[0];
        const float mm  = stats[1 + b];
        const float den = stats[33 + b];
        c = __expf(t * d_in[gid] - mm) / den;       // softmax over axis 1
    }

    float*       urow = uhat + (size_t)gid * N_DIM;
    const float* brow = bias + (size_t)n * N_DIM;

    v4f u[8], bv[8];
#pragma unroll
    for (int i = 0; i < 8; ++i) {
        u[i]  = ((const v4f*)urow)[lane + i * 32];
        bv[i] = ((const v4f*)brow)[lane + i * 32];
    }

    float sq = 0.f;
#pragma unroll
    for (int i = 0; i < 8; ++i) {
        v4f s = c * u[i] + bv[i];
        sq += s.x * s.x + s.y * s.y + s.z * s.z + s.w * s.w;
    }
#pragma unroll
    for (int m = 1; m < 32; m <<= 1) sq += __shfl_xor(sq, m, 32);

    const float f = sq / ((1.f + sq) * sqrtf(sq));

    if (mode == 2) {
#pragma unroll
        for (int i = 0; i < 8; ++i) {
            v4f s = c * u[i] + bv[i];
            ((v4f*)urow)[lane + i * 32] = f * s;    // in-place v_j
        }
    } else {
        float dq = 0.f;
#pragma unroll
        for (int i = 0; i < 8; ++i) {
            v4f s = c * u[i] + bv[i];
            v4f e = f * s - u[i];
            dq += e.x * e.x + e.y * e.y + e.z * e.z + e.w * e.w;
        }
#pragma unroll
        for (int m = 1; m < 32; m <<= 1) dq += __shfl_xor(dq, m, 32);
        if (lane == 0) d_new[gid] = sqrtf(dq);
    }
}

// ---------------------------------------------------------------------------
// Global reduction between routing passes:
//   d_o = mean(d);  t = t_num / (-0.5*d_o + eps)
//   per batch: mmax = max_n(t*d), denom = sum_n exp(t*d - mmax)
// One block, 32 waves; wave w handles batch w in phase B.
// ---------------------------------------------------------------------------
__global__ __launch_bounds__(1024) void caps_reduce(
    const float* __restrict__ d, float* __restrict__ stats)
{
    __shared__ float red[32];
    __shared__ float sh_t;
    const int tid  = threadIdx.x;
    const int lane = tid & 31;
    const int wave = tid >> 5;

    float s = 0.f;
    for (int i = tid; i < B_DIM * N_DIM; i += 1024) s += d[i];
#pragma unroll
    for (int m = 1; m < 32; m <<= 1) s += __shfl_xor(s, m, 32);
    if (lane == 0) red[wave] = s;
    __syncthreads();
    if (tid == 0) {
        float tot = 0.f;
        for (int i = 0; i < 32; ++i) tot += red[i];
        const float d_o = tot / (float)(B_DIM * N_DIM);
        // t_num = log(0.9*1023) - log(0.1) = log(9207)
        const float t = 9.1277390f / (-0.5f * d_o + 1e-12f);
        stats[0] = t;
        sh_t = t;
    }
    __syncthreads();

    const float t = sh_t;
    const int bb = wave;                      // 32 waves == 32 batches
    const float* db = d + bb * N_DIM;

    float mx = -3.4e38f;
    for (int i = lane; i < N_DIM; i += 32) mx = fmaxf(mx, t * db[i]);
#pragma unroll
    for (int m = 1; m < 32; m <<= 1) mx = fmaxf(mx, __shfl_xor(mx, m, 32));

    float se = 0.f;
    for (int i = lane; i < N_DIM; i += 32) se += __expf(t * db[i] - mx);
#pragma unroll
    for (int m = 1; m < 32; m <<= 1) se += __shfl_xor(se, m, 32);

    if (lane == 0) { stats[1 + bb] = mx; stats[33 + bb] = se; }
}

// ---------------------------------------------------------------------------
extern "C" void kernel_launch(void* const* d_in, const int* in_sizes, int n_in,
                              void* d_out, int out_size, void* d_ws, size_t ws_size,
                              hipStream_t stream)
{
    const float* x    = (const float*)d_in[0];   // (32,128,1024) f32
    const float* bias = (const float*)d_in[1];   // (1,1024,1024) f32
    float* out  = (float*)d_out;                 // (32,1024,1024) f32; doubles
                                                 // as u_hat storage (in-place)
    float* d_arr = (float*)d_ws;                 // 32*1024 floats
    float* stats = d_arr + B_DIM * N_DIM;        // [t, mmax[32], denom[32]]

    dim3 g1(N_DIM / 16, B_DIM);                  // 64 M-tiles x 32 batches
    caps_gemm_scale<<<g1, 256, 0, stream>>>(x, out);

    const int pass_blocks = (B_DIM * N_DIM) / 8; // 1 wave per row, 8 rows/block
    // iteration 0: c = 1/N
    caps_route_pass<<<pass_blocks, 256, 0, stream>>>(out, bias, d_arr, stats, d_arr, 0);
    caps_reduce<<<1, 1024, 0, stream>>>(d_arr, stats);
    // iteration 1
    caps_route_pass<<<pass_blocks, 256, 0, stream>>>(out, bias, d_arr, stats, d_arr, 1);
    caps_reduce<<<1, 1024, 0, stream>>>(d_arr, stats);
    // iteration 2: write v_j in place over u_hat
    caps_route_pass<<<pass_blocks, 256, 0, stream>>>(out, bias, d_arr, stats, d_arr, 2);
}